// gat_layer_13271448945165
// MI455X (gfx1250) — compile-verified
//
#include <hip/hip_runtime.h>

#define NFEAT 128
#define NHID 128
#define HEADS 2
#define FOUT 64
#define NEG_SLOPE 0.2f
#define LN_EPS 1e-5f

typedef __attribute__((ext_vector_type(2))) float v2f;
typedef __attribute__((ext_vector_type(8))) float v8f;

// ---------------------------------------------------------------------------
// feat = h @ W   ([n,128] x [128,128] -> [n,128]) via V_WMMA_F32_16X16X4_F32.
// One wave computes a full 16x128 output strip (8 accumulators): the A
// fragment is loaded once per k-step and reused by 8 WMMAs (8x less A
// traffic, 8 WMMAs per 18 loaded floats).  n is a multiple of 16 and all
// branches are wave-uniform, so EXEC is all-ones around every WMMA.
// fp32 A layout (16x4): lanes 0-15 hold M=0..15 K={k0,k0+1}; lanes 16-31 hold
// K={k0+2,k0+3}.  B (4x16): vgpr v, lanes 0-15 -> K=k0+v, lanes 16-31 ->
// K=k0+2+v, N=lane&15.  C (16x16): vgpr r, lanes 0-15 -> M=r, lanes16-31 M=r+8.
// ---------------------------------------------------------------------------
__global__ void gemm_feat_kernel(const float* __restrict__ h,
                                 const float* __restrict__ Wm,
                                 float* __restrict__ feat,
                                 int nstrips /* n/16 */) {
    const int lane  = threadIdx.x & 31;
    const int wave  = threadIdx.x >> 5;
    const int strip = blockIdx.x * 8 + wave;
    if (strip >= nstrips) return;               // wave-uniform

    const int m0  = strip << 4;                 // row tile * 16
    const int rA  = m0 + (lane & 15);
    const int kHi = (lane >> 4) << 1;           // 0 for lanes 0-15, 2 otherwise
    const int cB  = lane & 15;

    v8f acc[8];
#pragma unroll
    for (int t = 0; t < 8; ++t) acc[t] = (v8f){};

    const float* __restrict__ arow = h + (size_t)rA * NFEAT + kHi;
#pragma unroll 4
    for (int k0 = 0; k0 < NFEAT; k0 += 4) {
        v2f a;
        a.x = arow[k0];
        a.y = arow[k0 + 1];
        const float* __restrict__ brow0 = Wm + (size_t)(k0 + kHi) * NHID + cB;
        const float* __restrict__ brow1 = brow0 + NHID;
#pragma unroll
        for (int t = 0; t < 8; ++t) {
            v2f b;
            b.x = brow0[t * 16];
            b.y = brow1[t * 16];
            acc[t] = __builtin_amdgcn_wmma_f32_16x16x4_f32(
                false, a, false, b, (short)0, acc[t], false, false);
        }
    }

    const int rowBase = m0 + ((lane >> 4) << 3);
#pragma unroll
    for (int t = 0; t < 8; ++t) {
        float* __restrict__ ocol = feat + t * 16 + (lane & 15);
#pragma unroll
        for (int r = 0; r < 8; ++r)
            ocol[(size_t)(rowBase + r) * NHID] = acc[t][r];
    }
}

// ---------------------------------------------------------------------------
// el[n,h] = feat[n,h,:] . attn_l[h,:]   (and er with attn_r)
// One wave per (node, head); lane handles 2 consecutive floats; shfl reduce.
// ---------------------------------------------------------------------------
__global__ void attn_proj_kernel(const float* __restrict__ feat,
                                 const float* __restrict__ attn_l,
                                 const float* __restrict__ attn_r,
                                 float* __restrict__ el,
                                 float* __restrict__ er, int n) {
    const int w    = (int)((blockIdx.x * blockDim.x + threadIdx.x) >> 5);
    const int lane = threadIdx.x & 31;
    const int node = w >> 1;
    const int head = w & 1;
    if (node >= n) return;

    const int f0 = lane << 1;
    const float2 fv = *(const float2*)(feat + (size_t)node * NHID + head * FOUT + f0);
    const float2 al = *(const float2*)(attn_l + head * FOUT + f0);
    const float2 ar = *(const float2*)(attn_r + head * FOUT + f0);
    float sl = fv.x * al.x + fv.y * al.y;
    float sr = fv.x * ar.x + fv.y * ar.y;
#pragma unroll
    for (int m = 16; m >= 1; m >>= 1) {
        sl += __shfl_xor(sl, m, 32);
        sr += __shfl_xor(sr, m, 32);
    }
    if (lane == 0) {
        el[node * 2 + head] = sl;
        er[node * 2 + head] = sr;
    }
}

__global__ void init_emax_kernel(unsigned int* __restrict__ emax, int n2) {
    const int i = blockIdx.x * blockDim.x + threadIdx.x;
    if (i < n2) emax[i] = 0xFF800000u;   // -inf bit pattern
}

__device__ __forceinline__ void atomicMaxF(float* addr, float v) {
    // monotone bit trick: positives compare as ints, negatives reversed as uints
    if (v >= 0.0f) atomicMax((int*)addr, __float_as_int(v));
    else           atomicMin((unsigned int*)addr, __float_as_uint(v));
}

// ---------------------------------------------------------------------------
// Pass 1: e = leaky_relu(el[src] + er[dst]); emax[dst] = max(...)
// ---------------------------------------------------------------------------
__global__ void edge_logits_kernel(const int* __restrict__ src,
                                   const int* __restrict__ dst,
                                   const float* __restrict__ el,
                                   const float* __restrict__ er,
                                   float* __restrict__ ebuf,
                                   float* __restrict__ emax, int E) {
    const int i = blockIdx.x * blockDim.x + threadIdx.x;
    if (i >= E) return;
    const int s = src[i], d = dst[i];
    const float2 l = *(const float2*)(el + (size_t)s * 2);
    const float2 r = *(const float2*)(er + (size_t)d * 2);
    float e0 = l.x + r.x;  e0 = (e0 > 0.0f) ? e0 : e0 * NEG_SLOPE;
    float e1 = l.y + r.y;  e1 = (e1 > 0.0f) ? e1 : e1 * NEG_SLOPE;
    ebuf[(size_t)i * 2]     = e0;
    ebuf[(size_t)i * 2 + 1] = e1;
    atomicMaxF(emax + (size_t)d * 2,     e0);
    atomicMaxF(emax + (size_t)d * 2 + 1, e1);
}

// ---------------------------------------------------------------------------
// Pass 2: expe = exp(e - emax[dst]); denom[dst] += expe  (expe stored in-place)
// ---------------------------------------------------------------------------
__global__ void edge_softmax_kernel(const int* __restrict__ dst,
                                    float* __restrict__ ebuf,
                                    const float* __restrict__ emax,
                                    float* __restrict__ denom, int E) {
    const int i = blockIdx.x * blockDim.x + threadIdx.x;
    if (i >= E) return;
    const int d = dst[i];
    const float2 m = *(const float2*)(emax + (size_t)d * 2);
    const float x0 = __expf(ebuf[(size_t)i * 2]     - m.x);
    const float x1 = __expf(ebuf[(size_t)i * 2 + 1] - m.y);
    ebuf[(size_t)i * 2]     = x0;
    ebuf[(size_t)i * 2 + 1] = x1;
    atomicAdd(denom + (size_t)d * 2,     x0);
    atomicAdd(denom + (size_t)d * 2 + 1, x1);
}

// ---------------------------------------------------------------------------
// Pass 3 (heavy): out[dst] += alpha * feat[src].  One wave per edge, each lane
// owns a float4 slice of the 128-wide feature; lanes 0-15 = head 0.
// ---------------------------------------------------------------------------
__global__ void scatter_kernel(const int* __restrict__ src,
                               const int* __restrict__ dst,
                               const float* __restrict__ feat,
                               const float* __restrict__ ebuf,
                               const float* __restrict__ denom,
                               float* __restrict__ out, int E) {
    const int w    = (int)((blockIdx.x * blockDim.x + threadIdx.x) >> 5);
    const int lane = threadIdx.x & 31;
    if (w >= E) return;
    const int s = src[w], d = dst[w];
    const int head = lane >> 4;
    const float2 ee = *(const float2*)(ebuf  + (size_t)w * 2);
    const float2 dd = *(const float2*)(denom + (size_t)d * 2);
    const float alpha = (head ? ee.y : ee.x) / (head ? dd.y : dd.x);
    const float4 f = *(const float4*)(feat + (size_t)s * NHID + lane * 4);
    float* o = out + (size_t)d * NHID + lane * 4;
    atomicAdd(o + 0, f.x * alpha);
    atomicAdd(o + 1, f.y * alpha);
    atomicAdd(o + 2, f.z * alpha);
    atomicAdd(o + 3, f.w * alpha);
}

// ---------------------------------------------------------------------------
// LayerNorm over 128 features, in place on d_out.  One wave per node.
// ---------------------------------------------------------------------------
__global__ void layernorm_kernel(float* __restrict__ x,
                                 const float* __restrict__ bias,
                                 const float* __restrict__ gamma,
                                 const float* __restrict__ beta, int n) {
    const int w    = (int)((blockIdx.x * blockDim.x + threadIdx.x) >> 5);
    const int lane = threadIdx.x & 31;
    if (w >= n) return;

    float4 v = *(float4*)(x + (size_t)w * NHID + lane * 4);
    const float4 b = *(const float4*)(bias + lane * 4);
    v.x += b.x; v.y += b.y; v.z += b.z; v.w += b.w;

    float s = v.x + v.y + v.z + v.w;
#pragma unroll
    for (int m = 16; m >= 1; m >>= 1) s += __shfl_xor(s, m, 32);
    const float mu = s * (1.0f / NHID);

    const float dx = v.x - mu, dy = v.y - mu, dz = v.z - mu, dw = v.w - mu;
    float q = dx * dx + dy * dy + dz * dz + dw * dw;
#pragma unroll
    for (int m = 16; m >= 1; m >>= 1) q += __shfl_xor(q, m, 32);
    const float inv = rsqrtf(q * (1.0f / NHID) + LN_EPS);

    const float4 g  = *(const float4*)(gamma + lane * 4);
    const float4 bt = *(const float4*)(beta + lane * 4);
    float4 o;
    o.x = g.x * dx * inv + bt.x;
    o.y = g.y * dy * inv + bt.y;
    o.z = g.z * dz * inv + bt.z;
    o.w = g.w * dw * inv + bt.w;
    *(float4*)(x + (size_t)w * NHID + lane * 4) = o;
}

extern "C" void kernel_launch(void* const* d_in, const int* in_sizes, int n_in,
                              void* d_out, int out_size, void* d_ws, size_t ws_size,
                              hipStream_t stream) {
    const float* h      = (const float*)d_in[0];
    const int*   src    = (const int*)d_in[1];
    const int*   dst    = (const int*)d_in[2];
    const float* W      = (const float*)d_in[3];
    const float* attn_l = (const float*)d_in[4];
    const float* attn_r = (const float*)d_in[5];
    const float* bias   = (const float*)d_in[6];
    const float* gamma  = (const float*)d_in[7];
    const float* beta   = (const float*)d_in[8];
    float*       out    = (float*)d_out;

    const int n = in_sizes[0] / NFEAT;   // 100000
    const int E = in_sizes[1];           // 1600000

    // Workspace layout
    char* p = (char*)d_ws;
    float* feat  = (float*)p;  p += (size_t)n * NHID * sizeof(float);
    float* ebuf  = (float*)p;  p += (size_t)E * 2 * sizeof(float);
    float* el    = (float*)p;  p += (size_t)n * 2 * sizeof(float);
    float* er    = (float*)p;  p += (size_t)n * 2 * sizeof(float);
    float* emax  = (float*)p;  p += (size_t)n * 2 * sizeof(float);
    float* denom = (float*)p;  p += (size_t)n * 2 * sizeof(float);

    // Fresh state every call (graph-capture safe)
    hipMemsetAsync(out,   0, (size_t)n * NHID * sizeof(float), stream);
    hipMemsetAsync(denom, 0, (size_t)n * 2 * sizeof(float), stream);
    init_emax_kernel<<<(n * 2 + 255) / 256, 256, 0, stream>>>((unsigned int*)emax, n * 2);

    // Dense GEMM on the matrix pipe: one wave per 16x128 strip
    const int nstrips = n / 16;
    gemm_feat_kernel<<<(nstrips + 7) / 8, 256, 0, stream>>>(h, W, feat, nstrips);

    // el / er projections
    attn_proj_kernel<<<(n * 2 * 32 + 255) / 256, 256, 0, stream>>>(feat, attn_l, attn_r, el, er, n);

    // Edge softmax passes
    edge_logits_kernel<<<(E + 255) / 256, 256, 0, stream>>>(src, dst, el, er, ebuf, emax, E);
    edge_softmax_kernel<<<(E + 255) / 256, 256, 0, stream>>>(dst, ebuf, emax, denom, E);

    // Weighted message scatter (HBM-bound pass)
    scatter_kernel<<<(E + 7) / 8, 256, 0, stream>>>(src, dst, feat, ebuf, denom, out, E);

    // Fused bias + LayerNorm, in place on d_out
    layernorm_kernel<<<(n + 7) / 8, 256, 0, stream>>>(out, bias, gamma, beta, n);
}